// EdgePredictor_58007828300460
// MI455X (gfx1250) — compile-verified
//
#include <hip/hip_runtime.h>
#include <hip/hip_bf16.h>
#include <math.h>

// Problem constants from the reference
#define BB 16
#define NN 256
#define FF 32
#define H1DIM 64   // layer-1 width (= 2F)
#define H2DIM 32   // layer-2 width

typedef float v2f __attribute__((ext_vector_type(2)));
typedef float v4f __attribute__((ext_vector_type(4)));
typedef float v8f __attribute__((ext_vector_type(8)));

// WMMA-native permutation of a 64-wide hidden vector:
//   low-half  (lanes 0-15,  K = 4t+0, 4t+1) -> positions 2t, 2t+1        (0..31)
//   high-half (lanes 16-31, K = 4t+2, 4t+3) -> positions 32+2t, 32+2t+1  (32..63)
__device__ __forceinline__ int wmma_pos(int h) {
    const int t = h >> 2, r = h & 3;
    return (r < 2) ? (2 * t + r) : (32 + 2 * t + (r - 2));
}

// ---------------------------------------------------------------------------
// Kernel 1: per-node layer-1 factorization, stored in WMMA-permuted layout.
//   U[node][pos(h)] = W1[h][0:F] . x[node] + b1[h]
//   V[node][pos(h)] = W1[h][F:2F] . x[node]
// so that h1(i,j) = relu(U[i] + V[j]).  One block per node, 64 threads.
// ---------------------------------------------------------------------------
__global__ void ep_node_proj(const float* __restrict__ x,
                             const float* __restrict__ W1,
                             const float* __restrict__ b1,
                             float* __restrict__ U,
                             float* __restrict__ V) {
    __shared__ float xs[FF];
    const int node = blockIdx.x;      // 0 .. B*N-1
    const int h    = threadIdx.x;     // 0 .. 63
    if (h < FF) xs[h] = x[node * FF + h];
    __syncthreads();
    const float* wrow = W1 + h * (2 * FF);
    float u = 0.0f, v = 0.0f;
#pragma unroll
    for (int f = 0; f < FF; ++f) {
        u = fmaf(wrow[f],      xs[f], u);
        v = fmaf(wrow[FF + f], xs[f], v);
    }
    const int pos = wmma_pos(h);
    U[node * H1DIM + pos] = u + b1[h];
    V[node * H1DIM + pos] = v;
}

// ---------------------------------------------------------------------------
// Kernel 1b: permute W2 [32 x 64] rows into the same WMMA-native K order.
// ---------------------------------------------------------------------------
__global__ void ep_permute_w2(const float* __restrict__ W2,
                              float* __restrict__ Wp) {
    const int idx = blockIdx.x * blockDim.x + threadIdx.x;  // 0 .. 32*64-1
    const int k = idx >> 6;
    const int h = idx & 63;
    Wp[k * H1DIM + wmma_pos(h)] = W2[k * H1DIM + h];
}

// ---------------------------------------------------------------------------
// Kernel 2: layers 2+3 via V_WMMA_F32_16X16X4_F32.
// One wave32 per (batch b, row i); loops over 16 j-tiles of 16 pairs.
// All fragment data is in permuted layout: each lane's 32 floats are
// contiguous -> 8x global_load_b128 per fragment, issued in a batch.
// Next tile's V rows are prefetched (global_prefetch_b8).
// __launch_bounds__(128, 1): this kernel holds W2 (64 VGPRs), U (32), the
// current V tile (32) and two accumulators (16) resident -> needs ~170 VGPRs;
// trade occupancy for zero scratch spills (4096 waves still saturate the chip).
// ---------------------------------------------------------------------------
__global__ void __launch_bounds__(128, 1)
ep_pair_mlp(const float* __restrict__ U,
            const float* __restrict__ V,
            const float* __restrict__ Wp,
            const float* __restrict__ b2,
            const float* __restrict__ W3,
            const float* __restrict__ b3,
            float* __restrict__ S) {
    const int lane = threadIdx.x & 31;
    const int wid  = (blockIdx.x * blockDim.x + threadIdx.x) >> 5; // 0..B*N-1
    const int b    = wid >> 8;
    const int i    = wid & (NN - 1);
    const int p    = lane & 15;                   // pair-slot / channel-slot
    const int half = (lane < 16) ? 0 : 32;        // half-wave sub-row offset

    // B fragments: permuted W2, resident for the whole wave (contiguous).
    v4f w4[2][8];
#pragma unroll
    for (int kb = 0; kb < 2; ++kb) {
        const float* wrow = Wp + (kb * 16 + p) * H1DIM + half;
#pragma unroll
        for (int q = 0; q < 8; ++q)
            w4[kb][q] = *(const v4f*)(wrow + 4 * q);
    }
    const float bias0 = b2[p];
    const float bias1 = b2[16 + p];
    const float w3a   = W3[p];
    const float w3b   = W3[16 + p];
    const float b3v   = b3[0];

    // A-half from U[b][i] (contiguous), loaded once per wave.
    v4f u4[8];
    const float* urow = U + (b * NN + i) * H1DIM + half;
#pragma unroll
    for (int q = 0; q < 8; ++q)
        u4[q] = *(const v4f*)(urow + 4 * q);

    float* srow = S + (size_t)(b * NN + i) * NN;
    const float* vbase = V + (size_t)(b * NN) * H1DIM + half;

    for (int jt = 0; jt < 16; ++jt) {
        const int j0 = jt * 16;

        // Batch-load this tile's V rows (8 x b128, one wait), then prefetch
        // the next tile's rows into near caches while we compute.
        const float* vrow = vbase + (size_t)(j0 + p) * H1DIM;
        v4f v4r[8];
#pragma unroll
        for (int q = 0; q < 8; ++q)
            v4r[q] = *(const v4f*)(vrow + 4 * q);
        if (jt < 15)
            __builtin_prefetch(vrow + 16 * H1DIM, 0, 1);

        // Accumulators initialized with layer-2 bias (broadcast along M).
        v8f acc0, acc1;
#pragma unroll
        for (int r = 0; r < 8; ++r) { acc0[r] = bias0; acc1[r] = bias1; }

        // 32 WMMAs: 16 K-steps x 2 N-blocks; A built on the fly = relu(u+v).
#pragma unroll
        for (int t = 0; t < 16; ++t) {
            const int q = t >> 1, e = (t & 1) * 2;
            v2f a, bA, bB;
            a.x  = fmaxf(u4[q][e]     + v4r[q][e],     0.0f);
            a.y  = fmaxf(u4[q][e + 1] + v4r[q][e + 1], 0.0f);
            bA.x = w4[0][q][e];  bA.y = w4[0][q][e + 1];
            bB.x = w4[1][q][e];  bB.y = w4[1][q][e + 1];
            acc0 = __builtin_amdgcn_wmma_f32_16x16x4_f32(
                false, a, false, bA, (short)0, acc0, false, false);
            acc1 = __builtin_amdgcn_wmma_f32_16x16x4_f32(
                false, a, false, bB, (short)0, acc1, false, false);
        }

        // Layer 3: relu, dot with W3 across N (lanes), sigmoid, store score.
#pragma unroll
        for (int r = 0; r < 8; ++r) {
            float part = fmaxf(acc0[r], 0.0f) * w3a +
                         fmaxf(acc1[r], 0.0f) * w3b;
            part += __shfl_xor(part, 1);
            part += __shfl_xor(part, 2);
            part += __shfl_xor(part, 4);
            part += __shfl_xor(part, 8);   // halves reduced independently
            const float sc = 1.0f / (1.0f + __expf(-(part + b3v)));
            // lane 0 holds pair r, lane 16 holds pair r+8
            if (p == 0) srow[j0 + r + ((lane == 16) ? 8 : 0)] = sc;
        }
    }
}

// ---------------------------------------------------------------------------
// Kernel 3: mask + symmetrize:  out = pm ? 0.5*(S + S^T) : 0
// ---------------------------------------------------------------------------
__global__ void ep_finalize(const float* __restrict__ S,
                            const unsigned char* __restrict__ mask,
                            float* __restrict__ out) {
    const int idx = blockIdx.x * blockDim.x + threadIdx.x;  // 0 .. B*N*N-1
    const int b = idx >> 16;
    const int r = idx & 0xFFFF;
    const int i = r >> 8;
    const int j = r & (NN - 1);
    const bool pm = (mask[b * NN + i] != 0) && (mask[b * NN + j] != 0) && (i != j);
    float val = 0.0f;
    if (pm)
        val = 0.5f * (S[idx] + S[((size_t)(b * NN + j)) * NN + i]);
    out[idx] = val;
}

// ---------------------------------------------------------------------------
// Host launcher. Inputs (setup_inputs order):
//   0 node_features f32 [B,N,F]   1 node_masks bool [B,N]
//   2 W1 f32 [64,64]  3 b1 [64]   4 W2 [32,64]  5 b2 [32]
//   6 W3 [1,32]       7 b3 [1]
// Workspace: U (1MB) | V (1MB) | S (4MB) | Wp (8KB)
// ---------------------------------------------------------------------------
extern "C" void kernel_launch(void* const* d_in, const int* in_sizes, int n_in,
                              void* d_out, int out_size, void* d_ws, size_t ws_size,
                              hipStream_t stream) {
    const float*         x  = (const float*)d_in[0];
    const unsigned char* m  = (const unsigned char*)d_in[1];
    const float*         W1 = (const float*)d_in[2];
    const float*         b1 = (const float*)d_in[3];
    const float*         W2 = (const float*)d_in[4];
    const float*         b2 = (const float*)d_in[5];
    const float*         W3 = (const float*)d_in[6];
    const float*         b3 = (const float*)d_in[7];

    float* U  = (float*)d_ws;
    float* V  = U + (size_t)BB * NN * H1DIM;
    float* S  = V + (size_t)BB * NN * H1DIM;
    float* Wp = S + (size_t)BB * NN * NN;

    // 1) per-node layer-1 factorization (WMMA-permuted layout)
    ep_node_proj<<<BB * NN, H1DIM, 0, stream>>>(x, W1, b1, U, V);

    // 1b) permute W2 rows into WMMA-native K order
    ep_permute_w2<<<(H2DIM * H1DIM) / 256, 256, 0, stream>>>(W2, Wp);

    // 2) pairwise layers 2+3 on WMMA: one wave per (b, i) -> 4096 waves
    ep_pair_mlp<<<(BB * NN * 32) / 128, 128, 0, stream>>>(U, V, Wp, b2, W3, b3, S);

    // 3) mask + symmetrize into d_out
    ep_finalize<<<(BB * NN * NN) / 256, 256, 0, stream>>>(S, m, (float*)d_out);
}